// GraphConvolution_8375186227896
// MI455X (gfx1250) — compile-verified
//
#include <hip/hip_runtime.h>

// ---------------------------------------------------------------------------
// relu((x * drop_mask) @ weight @ support)  ==  relu((x * drop_mask) @ W2)
// with W2 = weight @ support precomputed once (128x128, 64KB).
// Memory-bound: 768MB HBM streaming vs 16.4 GFLOP -> fp32 WMMA is free.
// W2 is staged to LDS per block via the CDNA5 Tensor Data Mover (TENSORcnt).
// ---------------------------------------------------------------------------

typedef __attribute__((ext_vector_type(2))) float v2f;
typedef __attribute__((ext_vector_type(8))) float v8f;
typedef unsigned int u32x4 __attribute__((ext_vector_type(4)));
typedef int          i32x4 __attribute__((ext_vector_type(4)));
typedef int          i32x8 __attribute__((ext_vector_type(8)));

#define DIM    128
#define KSTEPS 32   // 128 / 4   (V_WMMA_F32_16X16X4_F32)
#define NTILES 8    // 128 / 16

__device__ __forceinline__ v8f wmma_f32_16x16x4(v2f a, v2f b, v8f c) {
  // 8 args: (neg_a, A, neg_b, B, c_mod, C, reuse_a, reuse_b)
  return __builtin_amdgcn_wmma_f32_16x16x4_f32(false, a, false, b, (short)0, c,
                                               false, false);
}

// ---------------------------------------------------------------------------
// Kernel 1: W2 = weight @ support, stored to d_ws in the WMMA B-fragment
// ("swizzled") layout:
//   w2s[(t*8 + n)*64 + L*2 + j] = W2[4t + 2*(L>=16) + j][16n + (L&15)]
// so kernel 2 can read B fragments as conflict-free ds_load_b64 per lane.
// ---------------------------------------------------------------------------
__global__ __launch_bounds__(256) void w2_swizzle_gemm(
    const float* __restrict__ weight, const float* __restrict__ support,
    float* __restrict__ w2s) {
  const int lane = threadIdx.x & 31;
  const int wave = threadIdx.x >> 5;
  const int ln   = lane & 15;
  const int half = lane >> 4;
  const int m0   = wave * 16;  // 8 waves cover 128 rows

  v8f acc[NTILES] = {};

  for (int t = 0; t < KSTEPS; ++t) {
    const int ka = t * 4 + half * 2;
    // A fragment (weight): lane<16 -> K=4t,4t+1 ; lane>=16 -> K=4t+2,4t+3
    v2f a;
    a.x = weight[(m0 + ln) * DIM + ka];
    a.y = weight[(m0 + ln) * DIM + ka + 1];
#pragma unroll
    for (int n = 0; n < NTILES; ++n) {
      v2f b;
      b.x = support[ka * DIM + n * 16 + ln];
      b.y = support[(ka + 1) * DIM + n * 16 + ln];
      acc[n] = wmma_f32_16x16x4(a, b, acc[n]);
    }
  }

  // Scatter D (16x16 f32 C/D layout) into the swizzled B-fragment layout.
#pragma unroll
  for (int n = 0; n < NTILES; ++n) {
#pragma unroll
    for (int r = 0; r < 8; ++r) {
      const int m  = m0 + r + half * 8;   // W2 row (= K index of main GEMM)
      const int t  = m >> 2;
      const int j  = m & 1;
      const int hb = (m >> 1) & 1;
      const int L  = hb * 16 + ln;        // column's lane slot (col&15 == ln)
      w2s[(t * NTILES + n) * 64 + L * 2 + j] = acc[n][r];
    }
  }
}

// ---------------------------------------------------------------------------
// Kernel 2: out = relu((x * mask) @ W2). One 16-row tile per wave, 8 waves
// per block. W2 (64KB, fragment-swizzled) staged into LDS once per block.
// ---------------------------------------------------------------------------
__global__ __launch_bounds__(256) void gcn_fused(
    const float* __restrict__ x, const float* __restrict__ mask,
    const float* __restrict__ w2s, float* __restrict__ out, int ntiles) {
  extern __shared__ float lds[];  // 16384 floats = 64KB
  const int tid  = threadIdx.x;
  const int lane = tid & 31;
  const int wave = tid >> 5;

#if __has_builtin(__builtin_amdgcn_tensor_load_to_lds) && \
    __has_builtin(__builtin_amdgcn_s_wait_tensorcnt)
  // --- Tensor Data Mover: one DMA op moves all 64KB of W2 into LDS. ---
  if (wave == 0) {
    const unsigned long long ga = (unsigned long long)(const void*)w2s;
    const unsigned lds_base =
        (unsigned)(unsigned long long)(__attribute__((address_space(3))) const void*)lds;
    // D# group 0: count=1 (valid user desc), lds_addr, global_addr, type=2.
    u32x4 g0 = {1u, lds_base, (unsigned)(ga & 0xFFFFFFFFu),
                (unsigned)((ga >> 32) & 0x01FFFFFFu) | (2u << 30)};
    // D# group 1: mask=0, data_size=4B; tensor 128x128, tile 128x128,
    // dim0 stride 128 (rows contiguous) -> flat 64KB linear move.
    i32x8 g1 = {(int)(2u << 16),    // [17:16] data_size = 2 (4 bytes)
                (int)(128u << 16),  // [63:48]  tensor_dim0 = 128
                (int)(128u << 16),  // [95:80]  tensor_dim1 = 128
                (int)(128u << 16),  // [127:112] tile_dim0  = 128
                128,                // [143:128] tile_dim1  = 128
                128,                // [191:160] tensor_dim0_stride = 128
                0, 0};
    i32x4 g2 = {0, 0, 0, 0};  // 2D tensor: groups 2/3 unused
    i32x4 g3 = {0, 0, 0, 0};
#if __has_include(<hip/amd_detail/amd_gfx1250_TDM.h>)
    i32x8 g4 = {0, 0, 0, 0, 0, 0, 0, 0};
    __builtin_amdgcn_tensor_load_to_lds(g0, g1, g2, g3, g4, 0);
#else
    __builtin_amdgcn_tensor_load_to_lds(g0, g1, g2, g3, 0);
#endif
    __builtin_amdgcn_s_wait_tensorcnt(0);
  }
#else
  // --- Fallback: flat coalesced copy (b128 loads/stores). ---
  {
    const float4* src4 = reinterpret_cast<const float4*>(w2s);
    float4*       dst4 = reinterpret_cast<float4*>(lds);
#pragma unroll
    for (int i = 0; i < 16; ++i) {
      const int idx = i * 256 + tid;
      dst4[idx] = src4[idx];
    }
  }
#endif
  __syncthreads();

  const int tile = blockIdx.x * 8 + wave;
  if (tile >= ntiles) return;

  const int  ln   = lane & 15;
  const int  half = lane >> 4;
  const long row  = (long)tile * 16 + ln;
  // A-fragment base: this lane's row, starting at K sub-offset (half*2)
  const float* xr = x    + row * DIM + half * 2;
  const float* mr = mask + row * DIM + half * 2;

  v8f acc[NTILES] = {};

#pragma unroll 4
  for (int t = 0; t < KSTEPS; ++t) {
    // Streaming data: non-temporal loads (touched exactly once).
    v2f xa = __builtin_nontemporal_load(reinterpret_cast<const v2f*>(xr + t * 4));
    v2f ma = __builtin_nontemporal_load(reinterpret_cast<const v2f*>(mr + t * 4));
    v2f a  = xa * ma;  // inverted-dropout applied in registers
#pragma unroll
    for (int n = 0; n < NTILES; ++n) {
      // conflict-free ds_load_b64: lane L reads 8*L within the fragment
      v2f b = *reinterpret_cast<const v2f*>(
          &lds[(t * NTILES + n) * 64 + lane * 2]);
      acc[n] = wmma_f32_16x16x4(a, b, acc[n]);
    }
  }

  // Epilogue: ReLU + row-major non-temporal store (streamed output).
  float* orow = out + (long)tile * 16 * DIM;
#pragma unroll
  for (int n = 0; n < NTILES; ++n) {
#pragma unroll
    for (int r = 0; r < 8; ++r) {
      __builtin_nontemporal_store(fmaxf(acc[n][r], 0.0f),
                                  &orow[(r + half * 8) * DIM + n * 16 + ln]);
    }
  }
}

// ---------------------------------------------------------------------------
extern "C" void kernel_launch(void* const* d_in, const int* in_sizes, int n_in,
                              void* d_out, int out_size, void* d_ws,
                              size_t ws_size, hipStream_t stream) {
  const float* x       = (const float*)d_in[0];
  const float* weight  = (const float*)d_in[1];
  const float* support = (const float*)d_in[2];
  const float* drop    = (const float*)d_in[3];
  float*       out     = (float*)d_out;
  float*       w2s     = (float*)d_ws;  // 128*128 floats = 64KB scratch

  const int nrows   = in_sizes[0] / DIM;  // 500000
  const int ntiles  = nrows / 16;         // 31250 (N % 16 == 0)
  const int nblocks = (ntiles + 7) / 8;   // 8 waves (tiles) per block

  w2_swizzle_gemm<<<1, 256, 0, stream>>>(weight, support, w2s);
  gcn_fused<<<nblocks, 256, 65536, stream>>>(x, drop, w2s, out, ntiles);
}